// fusion_sf_19172734009977
// MI455X (gfx1250) — compile-verified
//
#include <hip/hip_runtime.h>

// Problem constants from the reference: (B,C,H,W) = (4,32,512,512), radius 5.
#define BB  4
#define CC  32
#define HH  512
#define WW  512

typedef __attribute__((ext_vector_type(2))) float v2f;  // WMMA f32 16x16x4 A/B: 2 VGPRs
typedef __attribute__((ext_vector_type(8))) float v8f;  // WMMA f32 16x16 C/D: 8 VGPRs

// -----------------------------------------------------------------------------
// Kernel 1: channel-summed squared gradient for BOTH inputs in one pass.
// This is the bandwidth-dominant pass (reads 256 MiB, writes 8 MiB) -> keep it
// as plain coalesced f32 loads; the 192 MB L2 absorbs the (i-1,j) / (i,j-1)
// re-reads. Emits global_prefetch_b8 for the next channel plane.
// -----------------------------------------------------------------------------
__global__ void grad_csum_kernel(const float* __restrict__ f1,
                                 const float* __restrict__ f2,
                                 float* __restrict__ g1,
                                 float* __restrict__ g2) {
  const int idx = blockIdx.x * blockDim.x + threadIdx.x;   // 0 .. B*H*W-1
  const int j = idx & (WW - 1);
  const int i = (idx / WW) & (HH - 1);
  const int b = idx / (WW * HH);
  const int base = ((b * CC) * HH + i) * WW + j;           // channel 0 offset
  const int cstride = HH * WW;

  float acc1 = 0.0f, acc2 = 0.0f;
#pragma unroll 4
  for (int c = 0; c < CC; ++c) {
    const int off = base + c * cstride;
    if (c + 1 < CC) {
      __builtin_prefetch(f1 + off + cstride, 0, 1);  // -> global_prefetch_b8
      __builtin_prefetch(f2 + off + cstride, 0, 1);
    }
    const float x1 = f1[off];
    const float l1 = (j > 0) ? f1[off - 1] : 0.0f;    // r_shift = f[i, j-1]
    const float t1 = (i > 0) ? f1[off - WW] : 0.0f;   // b_shift = f[i-1, j]
    acc1 += (l1 - x1) * (l1 - x1) + (t1 - x1) * (t1 - x1);

    const float x2 = f2[off];
    const float l2 = (j > 0) ? f2[off - 1] : 0.0f;
    const float t2 = (i > 0) ? f2[off - WW] : 0.0f;
    acc2 += (l2 - x2) * (l2 - x2) + (t2 - x2) * (t2 - x2);
  }
  g1[idx] = acc1;
  g2[idx] = acc2;
}

// Band matrix value for an 11-wide box spanning 3 aligned 16-tiles.
//   T = -1 (prev tile): 1 iff k >= r+11
//   T =  0 (this tile): 1 iff |k - r| <= 5
//   T = +1 (next tile): 1 iff k <= r-11
__device__ __forceinline__ float band_val(int T, int r, int k) {
  if (T < 0) return (k >= r + 11) ? 1.0f : 0.0f;
  if (T > 0) return (k <= r - 11) ? 1.0f : 0.0f;
  const int d = k - r;
  return (d >= -5 && d <= 5) ? 1.0f : 0.0f;
}

// -----------------------------------------------------------------------------
// Kernel 2: horizontal 11-tap box sum via V_WMMA_F32_16X16X4_F32.
// One wave per 16x16 tile:  D = sum_T  G_T(16x16) x Band_T^T(16x16),
// decomposed into 4 K-chunks of 4.  A[m,kk] = G[row m, col 4q+kk],
// B[kk,n] = Band[n, 4q+kk].  Both images processed with shared bands.
// -----------------------------------------------------------------------------
__global__ void hbox_wmma_kernel(const float* __restrict__ g1,
                                 const float* __restrict__ g2,
                                 float* __restrict__ h1,
                                 float* __restrict__ h2) {
  const int lane = threadIdx.x;        // 0..31, one wave
  const int nn   = lane & 15;
  const int half = lane >> 4;
  const int j0 = blockIdx.x * 16;
  const int i0 = blockIdx.y * 16;
  const int b  = blockIdx.z;

  const float* p1 = g1 + (size_t)b * HH * WW;
  const float* p2 = g2 + (size_t)b * HH * WW;

  v8f c1 = {}; v8f c2 = {};
#pragma unroll
  for (int T = -1; T <= 1; ++T) {
    const int jt = j0 + T * 16;                 // tile base column
    if (jt < 0 || jt >= WW) continue;           // block-uniform: EXEC stays full
#pragma unroll
    for (int q = 0; q < 4; ++q) {
      const int kk = q * 4 + half * 2;          // this lane's K pair within chunk
      // A-matrix layout: lane holds row (lane&15), K columns kk, kk+1
      const float* a1p = p1 + (size_t)(i0 + nn) * WW + jt + kk;
      const float* a2p = p2 + (size_t)(i0 + nn) * WW + jt + kk;
      v2f a1; a1.x = a1p[0]; a1.y = a1p[1];
      v2f a2; a2.x = a2p[0]; a2.y = a2p[1];
      // B-matrix layout: lane holds col (lane&15), K rows kk, kk+1
      v2f bm; bm.x = band_val(T, nn, kk); bm.y = band_val(T, nn, kk + 1);
      c1 = __builtin_amdgcn_wmma_f32_16x16x4_f32(false, a1, false, bm,
                                                 (short)0, c1, false, false);
      c2 = __builtin_amdgcn_wmma_f32_16x16x4_f32(false, a2, false, bm,
                                                 (short)0, c2, false, false);
    }
  }
  // C/D layout: lane holds col (lane&15); VGPR v holds row v + 8*(lane>>4).
  float* o1 = h1 + ((size_t)b * HH + i0 + 8 * half) * WW + j0 + nn;
  float* o2 = h2 + ((size_t)b * HH + i0 + 8 * half) * WW + j0 + nn;
#pragma unroll
  for (int v = 0; v < 8; ++v) {
    o1[(size_t)v * WW] = c1[v];
    o2[(size_t)v * WW] = c2[v];
  }
}

// -----------------------------------------------------------------------------
// Kernel 3: vertical 11-tap box sum via WMMA + final compare.
//   D = sum_T Band_T(16x16) x H_T(16x16);  out = (sf1 < sf2) ? 1 : 0
// A[m,kk] = Band[m, 4q+kk] (band on the A side now),
// B[kk,n] = H[tile row 4q+kk, col n].
// -----------------------------------------------------------------------------
__global__ void vbox_cmp_wmma_kernel(const float* __restrict__ h1,
                                     const float* __restrict__ h2,
                                     float* __restrict__ out) {
  const int lane = threadIdx.x;
  const int nn   = lane & 15;
  const int half = lane >> 4;
  const int j0 = blockIdx.x * 16;
  const int i0 = blockIdx.y * 16;
  const int b  = blockIdx.z;

  const float* p1 = h1 + (size_t)b * HH * WW;
  const float* p2 = h2 + (size_t)b * HH * WW;

  v8f c1 = {}; v8f c2 = {};
#pragma unroll
  for (int T = -1; T <= 1; ++T) {
    const int it = i0 + T * 16;                 // tile base row
    if (it < 0 || it >= HH) continue;           // block-uniform
#pragma unroll
    for (int q = 0; q < 4; ++q) {
      const int kk = q * 4 + half * 2;
      // A: band matrix, lane holds row (lane&15), K cols kk, kk+1
      v2f am; am.x = band_val(T, nn, kk); am.y = band_val(T, nn, kk + 1);
      // B: H tile, lane holds col (lane&15), K rows kk, kk+1 (stride W apart)
      const float* b1p = p1 + (size_t)(it + kk) * WW + j0 + nn;
      const float* b2p = p2 + (size_t)(it + kk) * WW + j0 + nn;
      v2f b1; b1.x = b1p[0]; b1.y = b1p[WW];
      v2f b2; b2.x = b2p[0]; b2.y = b2p[WW];
      c1 = __builtin_amdgcn_wmma_f32_16x16x4_f32(false, am, false, b1,
                                                 (short)0, c1, false, false);
      c2 = __builtin_amdgcn_wmma_f32_16x16x4_f32(false, am, false, b2,
                                                 (short)0, c2, false, false);
    }
  }
  float* o = out + ((size_t)b * HH + i0 + 8 * half) * WW + j0 + nn;
#pragma unroll
  for (int v = 0; v < 8; ++v) {
    o[(size_t)v * WW] = (c1[v] < c2[v]) ? 1.0f : 0.0f;
  }
}

// -----------------------------------------------------------------------------
extern "C" void kernel_launch(void* const* d_in, const int* in_sizes, int n_in,
                              void* d_out, int out_size, void* d_ws, size_t ws_size,
                              hipStream_t stream) {
  const float* f1 = (const float*)d_in[0];
  const float* f2 = (const float*)d_in[1];
  float* outp = (float*)d_out;

  const size_t NB = (size_t)BB * HH * WW;   // 1,048,576 elements (4 MiB)
  float* g1 = (float*)d_ws;                 // workspace: g1,g2,h1,h2 = 16 MiB
  float* g2 = g1 + NB;
  float* h1 = g2 + NB;
  float* h2 = h1 + NB;

  // Pass 1: gradient + channel reduction (bandwidth-bound, ~256 MiB read).
  grad_csum_kernel<<<(unsigned)(NB / 256), 256, 0, stream>>>(f1, f2, g1, g2);

  // Passes 2/3: separable 11x11 box via f32 WMMA band-matrix multiplies,
  // one wave per 16x16 tile, then final compare.
  dim3 grid(WW / 16, HH / 16, BB);
  hbox_wmma_kernel<<<grid, 32, 0, stream>>>(g1, g2, h1, h2);
  vbox_cmp_wmma_kernel<<<grid, 32, 0, stream>>>(h1, h2, outp);
}